// MPNN_enn_edge_15882789061280
// MI455X (gfx1250) — compile-verified
//
#include <hip/hip_runtime.h>
#include <hip/hip_bf16.h>

// ---------------------------------------------------------------------------
// MPNN (edge-network message passing + GRUCell), T=8 steps.
// N=10000 nodes, E=25000 edges, H=128.
//
// Bottleneck: streaming edge_data (1.64 GB) per step from HBM (23.3 TB/s).
// Edge kernel = pure VALU f32 streaming + f32 global atomics.
// GRU kernel  = v_wmma_f32_16x16x4_f32 tiles for gi/gh GEMMs + fused gates.
// ---------------------------------------------------------------------------

typedef __attribute__((ext_vector_type(2))) float v2f;  // A/B frag: 16x4 / 4x16 f32 = 2 VGPRs
typedef __attribute__((ext_vector_type(8))) float v8f;  // C/D: 16x16 f32 = 8 VGPRs

#define HD 128          // hidden dim
#define TSTEPS 8

// ---------------------------------------------------------------------------
__global__ void zero_kernel(float* __restrict__ p, int n) {
  int i = blockIdx.x * blockDim.x + threadIdx.x;
  if (i < n) p[i] = 0.0f;
}

// ---------------------------------------------------------------------------
// One block (128 threads) per edge.
// thread i computes edge_msg[e][i] = dot(edge_data[e][i][:], h[Esrc[e]][:])
// and atomically accumulates into node_msg[Etgt[e]][i].
// edge_data is read exactly once, contiguously per row -> streams HBM.
// ---------------------------------------------------------------------------
__global__ __launch_bounds__(HD) void edge_msg_kernel(
    const float* __restrict__ h,
    const int*  __restrict__ Esrc,
    const int*  __restrict__ Etgt,
    const float* __restrict__ edge_data,
    float* __restrict__ node_msg,
    int E)
{
  const int e   = blockIdx.x;
  const int tid = threadIdx.x;          // 0..127 = output row
  if (e >= E) return;

  __shared__ float xs[HD];
  const int s = Esrc[e];
  const int t = Etgt[e];
  xs[tid] = h[(size_t)s * HD + tid];
  __syncthreads();

  const float* row = edge_data + (size_t)e * HD * HD + (size_t)tid * HD;

  // speculative prefetch of the next edge's matching row (global_prefetch_b8)
  if (e + 1 < E)
    __builtin_prefetch(edge_data + (size_t)(e + 1) * HD * HD + (size_t)tid * HD, 0, 1);

  const float4* row4 = (const float4*)row;
  float acc = 0.0f;
#pragma unroll 8
  for (int j = 0; j < HD / 4; ++j) {
    float4 m = row4[j];
    acc += m.x * xs[4 * j + 0];
    acc += m.y * xs[4 * j + 1];
    acc += m.z * xs[4 * j + 2];
    acc += m.w * xs[4 * j + 3];
  }

  // scatter-sum (segment_sum over Etgt): global_atomic_add_f32, device scope
  __hip_atomic_fetch_add(&node_msg[(size_t)t * HD + tid], acc,
                         __ATOMIC_RELAXED, __HIP_MEMORY_SCOPE_AGENT);
}

// ---------------------------------------------------------------------------
// GRU cell with WMMA f32 16x16x4.
//   inp = concat([h, msg])                [N, 2H]
//   gi  = inp @ W_ih^T + b_ih            [N, 3H]
//   gh  = h   @ W_hh^T + b_hh            [N, 3H]
//   r = sigmoid(gi_r + gh_r); z = sigmoid(gi_z + gh_z)
//   n = tanh(gi_n + r * gh_n); h' = (1-z)*n + z*h
//
// One wave per (16-node tile m0, 16-column tile c0). The 3 gates for the
// same hidden columns live at 3H offsets c0, c0+128, c0+256 -> 6 v8f accums.
//
// f32 A-matrix 16x4 layout (ISA 7.12.2): lanes 0-15 hold K=0,1; lanes 16-31
// hold K=2,3; M = lane%16. B (4x16) mirrored: N = lane%16, same K split.
// C/D 16x16: d[j] = D[M = j + 8*(lane/16)][N = lane%16].
// ---------------------------------------------------------------------------
__global__ __launch_bounds__(32) void gru_wmma_kernel(
    const float* __restrict__ h,
    const float* __restrict__ msg,
    const float* __restrict__ W_ih,   // [3H, 2H] row-major
    const float* __restrict__ W_hh,   // [3H,  H] row-major
    const float* __restrict__ b_ih,   // [3H]
    const float* __restrict__ b_hh,   // [3H]
    float* __restrict__ h_out,
    int N)
{
  const int m0   = blockIdx.x * 16;   // node tile    (N % 16 == 0)
  const int c0   = blockIdx.y * 16;   // hidden-col tile
  const int lane = threadIdx.x;       // wave32
  const int nl    = lane & 15;
  const int khalf = lane >> 4;        // 0 or 1

  const int arow = m0 + nl;           // A-fragment row for this lane

  v8f air = (v8f)0.0f, aiz = (v8f)0.0f, ain = (v8f)0.0f;  // gi accum (r,z,n)
  v8f ahr = (v8f)0.0f, ahz = (v8f)0.0f, ahn = (v8f)0.0f;  // gh accum (r,z,n)

  // ---- gi: K = 2H over inp = [h | msg], W_ih stride 2H ----
  for (int k0 = 0; k0 < 2 * HD; k0 += 4) {
    const int kk = k0 + 2 * khalf;
    const float* asrc = (kk < HD) ? (h   + (size_t)arow * HD + kk)
                                  : (msg + (size_t)arow * HD + (kk - HD));
    v2f a; a.x = asrc[0]; a.y = asrc[1];

    const float* wr = W_ih + (size_t)(c0 +          nl) * (2 * HD) + kk;
    const float* wz = W_ih + (size_t)(c0 +     HD + nl) * (2 * HD) + kk;
    const float* wn = W_ih + (size_t)(c0 + 2 * HD + nl) * (2 * HD) + kk;
    v2f br; br.x = wr[0]; br.y = wr[1];
    v2f bz; bz.x = wz[0]; bz.y = wz[1];
    v2f bn; bn.x = wn[0]; bn.y = wn[1];

    air = __builtin_amdgcn_wmma_f32_16x16x4_f32(false, a, false, br, (short)0, air, false, false);
    aiz = __builtin_amdgcn_wmma_f32_16x16x4_f32(false, a, false, bz, (short)0, aiz, false, false);
    ain = __builtin_amdgcn_wmma_f32_16x16x4_f32(false, a, false, bn, (short)0, ain, false, false);
  }

  // ---- gh: K = H over h, W_hh stride H ----
  for (int k0 = 0; k0 < HD; k0 += 4) {
    const int kk = k0 + 2 * khalf;
    const float* asrc = h + (size_t)arow * HD + kk;
    v2f a; a.x = asrc[0]; a.y = asrc[1];

    const float* wr = W_hh + (size_t)(c0 +          nl) * HD + kk;
    const float* wz = W_hh + (size_t)(c0 +     HD + nl) * HD + kk;
    const float* wn = W_hh + (size_t)(c0 + 2 * HD + nl) * HD + kk;
    v2f br; br.x = wr[0]; br.y = wr[1];
    v2f bz; bz.x = wz[0]; bz.y = wz[1];
    v2f bn; bn.x = wn[0]; bn.y = wn[1];

    ahr = __builtin_amdgcn_wmma_f32_16x16x4_f32(false, a, false, br, (short)0, ahr, false, false);
    ahz = __builtin_amdgcn_wmma_f32_16x16x4_f32(false, a, false, bz, (short)0, ahz, false, false);
    ahn = __builtin_amdgcn_wmma_f32_16x16x4_f32(false, a, false, bn, (short)0, ahn, false, false);
  }

  // ---- gates + update (fused epilogue) ----
  const int c    = c0 + nl;
  const float bir = b_ih[c],          bhr = b_hh[c];
  const float biz = b_ih[c + HD],     bhz = b_hh[c + HD];
  const float bin = b_ih[c + 2 * HD], bhn = b_hh[c + 2 * HD];

#pragma unroll
  for (int j = 0; j < 8; ++j) {
    const int node = m0 + j + 8 * khalf;
    const float hv = h[(size_t)node * HD + c];
    const float rv = 1.0f / (1.0f + __expf(-(air[j] + bir + ahr[j] + bhr)));
    const float zv = 1.0f / (1.0f + __expf(-(aiz[j] + biz + ahz[j] + bhz)));
    const float nv = tanhf(ain[j] + bin + rv * (ahn[j] + bhn));
    h_out[(size_t)node * HD + c] = (1.0f - zv) * nv + zv * hv;
  }
}

// ---------------------------------------------------------------------------
extern "C" void kernel_launch(void* const* d_in, const int* in_sizes, int n_in,
                              void* d_out, int out_size, void* d_ws, size_t ws_size,
                              hipStream_t stream) {
  const float* x         = (const float*)d_in[0];
  const int*   Esrc      = (const int*)  d_in[1];
  const int*   Etgt      = (const int*)  d_in[2];
  const float* edge_data = (const float*)d_in[3];
  const float* W_ih      = (const float*)d_in[4];
  const float* W_hh      = (const float*)d_in[5];
  const float* b_ih      = (const float*)d_in[6];
  const float* b_hh      = (const float*)d_in[7];
  float* out = (float*)d_out;

  const int N = in_sizes[0] / HD;   // 10000
  const int E = in_sizes[1];        // 25000
  const size_t NH = (size_t)N * HD;

  float* msg   = (float*)d_ws;      // [N, H]
  float* hbuf0 = msg + NH;          // ping
  float* hbuf1 = hbuf0 + NH;        // pong

  const float* hprev = x;
  for (int t = 0; t < TSTEPS; ++t) {
    float* hnext = (t == TSTEPS - 1) ? out : ((t & 1) ? hbuf1 : hbuf0);

    zero_kernel<<<dim3((int)((NH + 255) / 256)), dim3(256), 0, stream>>>(msg, (int)NH);

    edge_msg_kernel<<<dim3(E), dim3(HD), 0, stream>>>(
        hprev, Esrc, Etgt, edge_data, msg, E);

    gru_wmma_kernel<<<dim3(N / 16, HD / 16), dim3(32), 0, stream>>>(
        hprev, msg, W_ih, W_hh, b_ih, b_hh, hnext, N);

    hprev = hnext;
  }
}